// TransformedInput_26998164423199
// MI455X (gfx1250) — compile-verified
//
#include <hip/hip_runtime.h>

// ---------------------------------------------------------------------------
// TransformedInput zonotope builder for MI455X (gfx1250).
//
// Output = (1+N, N) f32, N = 12288  ->  604 MB, >99.99% zeros.
// Bottleneck: streaming zero-fill at HBM bandwidth (23.3 TB/s -> ~26 us).
// Kernel 1: b128 streaming zero-fill using the CDNA5 async
//           GLOBAL_STORE_ASYNC_FROM_LDS_B128 path (ASYNCcnt), with a
//           non-temporal b128 store fallback.
// Kernel 2: single workgroup: compute center/err, block-wide inclusive scan
//           (stream compaction row index), write row 0 + scattered diagonal.
// ---------------------------------------------------------------------------

typedef __attribute__((ext_vector_type(4))) float v4f;
typedef __attribute__((ext_vector_type(4))) int   v4i;

#ifndef __has_builtin
#define __has_builtin(x) 0
#endif

#if __has_builtin(__builtin_amdgcn_global_store_async_from_lds_b128) && \
    __has_builtin(__builtin_amdgcn_s_wait_asynccnt)
#define USE_ASYNC_FILL 1
#else
#define USE_ASYNC_FILL 0
#endif

#define FILL_BLK 256
#define SCAN_BLK 1024
#define EPS_F 0.1f

// ---------------------------------------------------------------------------
// Kernel 1: zero the entire output buffer (n floats), 16B per lane per op.
// ---------------------------------------------------------------------------
__global__ void zono_fill_zero(float* __restrict__ out, long long n) {
  const long long nv = n >> 2;  // number of aligned float4 chunks
  const long long stride = (long long)gridDim.x * FILL_BLK;
  long long i = (long long)blockIdx.x * FILL_BLK + threadIdx.x;

#if USE_ASYNC_FILL
  // 4 KB LDS staging buffer of zeros; each lane owns a private 16B slot so
  // the async engine reads conflict-free b128 rows out of LDS.
  __shared__ alignas(16) float zbuf[FILL_BLK * 4];
  {
    v4f z = {0.0f, 0.0f, 0.0f, 0.0f};
    ((v4f*)zbuf)[threadIdx.x] = z;
  }
  __syncthreads();
  v4i* lsrc = (v4i*)zbuf + threadIdx.x;
  for (; i < nv; i += stride) {
    // copies 16 bytes from the LDS slot at lsrc to the i-th 16-byte chunk of out, tracked by ASYNCcnt
    __builtin_amdgcn_global_store_async_from_lds_b128(
        (v4i*)out + i, lsrc, /*offset=*/0, /*cpol=*/0);
  }
  __builtin_amdgcn_s_wait_asynccnt(0);
#else
  const v4f z = {0.0f, 0.0f, 0.0f, 0.0f};
  for (; i < nv; i += stride) {
    // Output (604MB) >> L2 (192MB): non-temporal streaming stores.
    __builtin_nontemporal_store(z, (v4f*)out + i);
  }
#endif

  // Scalar tail (out_size % 4 elements; zero for this problem, kept generic).
  for (long long e = (nv << 2) + (long long)blockIdx.x * FILL_BLK + threadIdx.x;
       e < n; e += stride) {
    out[e] = 0.0f;
  }
}

// ---------------------------------------------------------------------------
// Kernel 2: one block of 1024 threads.
//   lo  = relu(EPS - x) * 0.5
//   hi  = relu(x - (1-EPS)) * 0.5
//   center = x + lo - hi          -> out[0*N + k]
//   err = EPS - lo - hi
//   cond = err >= 0; row = inclusive-cumsum(cond)   (1-based)
//   if cond: out[row*N + k] = err
// ---------------------------------------------------------------------------
__global__ void zono_center_scatter(const float* __restrict__ x,
                                    float* __restrict__ out, int N) {
  const int t = threadIdx.x;
  const int per = (N + SCAN_BLK - 1) / SCAN_BLK;  // 12 for N=12288
  const int base = t * per;

  // Pass 1: count selected elements in this thread's contiguous chunk.
  int cnt = 0;
  for (int ii = 0; ii < per; ++ii) {
    const int k = base + ii;
    if (k < N) {
      const float xv = x[k];
      const float lo = fmaxf(EPS_F - xv, 0.0f) * 0.5f;
      const float hi = fmaxf(xv - (1.0f - EPS_F), 0.0f) * 0.5f;
      const float err = EPS_F - lo - hi;
      if (err >= 0.0f) ++cnt;
    }
  }

  // Block-wide inclusive scan over per-thread counts (Hillis-Steele in LDS).
  __shared__ int s[SCAN_BLK];
  s[t] = cnt;
  __syncthreads();
  for (int off = 1; off < SCAN_BLK; off <<= 1) {
    const int v = s[t];
    const int add = (t >= off) ? s[t - off] : 0;
    __syncthreads();
    s[t] = v + add;
    __syncthreads();
  }
  int run = s[t] - cnt;  // exclusive prefix for this thread's chunk

  // Pass 2: write center row and scattered error diagonal.
  for (int ii = 0; ii < per; ++ii) {
    const int k = base + ii;
    if (k < N) {
      const float xv = x[k];
      const float lo = fmaxf(EPS_F - xv, 0.0f) * 0.5f;
      const float hi = fmaxf(xv - (1.0f - EPS_F), 0.0f) * 0.5f;
      const float center = xv + lo - hi;
      const float err = EPS_F - lo - hi;
      out[k] = center;  // row 0
      if (err >= 0.0f) {
        ++run;  // inclusive cumsum value at k (1-based row)
        out[(long long)run * N + k] = err;
      }
    }
  }
}

// ---------------------------------------------------------------------------
// Host launcher
// ---------------------------------------------------------------------------
extern "C" void kernel_launch(void* const* d_in, const int* in_sizes, int n_in,
                              void* d_out, int out_size, void* d_ws,
                              size_t ws_size, hipStream_t stream) {
  (void)n_in; (void)d_ws; (void)ws_size;
  const float* x = (const float*)d_in[0];
  float* out = (float*)d_out;
  const int N = in_sizes[0];  // C*H*W = 12288
  const long long total = (long long)out_size;  // (1+N)*N floats

  const long long nv = total >> 2;
  long long b = (nv + FILL_BLK - 1) / FILL_BLK;
  if (b > 4096) b = 4096;
  if (b < 1) b = 1;

  zono_fill_zero<<<(int)b, FILL_BLK, 0, stream>>>(out, total);
  zono_center_scatter<<<1, SCAN_BLK, 0, stream>>>(x, out, N);
}